// GraphNetwork_19877108646549
// MI455X (gfx1250) — compile-verified
//
#include <hip/hip_runtime.h>
#include <hip/hip_bf16.h>

#define N_NODES 100000
#define N_EDGES 1000000
#define DD 64
#define HH 128

typedef float v2f __attribute__((ext_vector_type(2)));
typedef float v8f __attribute__((ext_vector_type(8)));

__device__ __forceinline__ v8f wmma_f32(v2f a, v2f b, v8f c) {
  // D = A(16x4 f32) x B(4x16 f32) + C(16x16 f32)
  return __builtin_amdgcn_wmma_f32_16x16x4_f32(
      /*neg_a=*/false, a, /*neg_b=*/false, b,
      /*c_mod=*/(short)0, c, /*reuse_a=*/false, /*reuse_b=*/false);
}

// Pair-interleaved, bank-swizzled weight layout in LDS:
//   element W[k][c] (C columns) -> word (k>>1)*(2C) + ((2c) ^ (32*((k>>1)&1))) + (k&1)
// B-fragment (k0,k0+1 at column c) is then ONE 8-byte-aligned ds_load_b64, and the
// XOR spreads half-waves onto disjoint bank halves (conflict-free b64).
__device__ __forceinline__ int wswz(int k, int c, int twoC) {
  const int row = k >> 1;
  return row * twoC + (((c << 1) ^ ((row & 1) << 5))) + (k & 1);
}
// hbuf swizzle: h[r][c] (64 cols) -> word r*64 + (c ^ (4*r)); keeps (c,c+1) pairs
// contiguous (XOR is a multiple of 4) and makes layer-2 A reads conflict-free.
__device__ __forceinline__ int hswz(int r, int c) { return r * 64 + (c ^ (r << 2)); }

// ---------------------------------------------------------------------------
// Zero agg[N*64] + sum_n[64] + sum_e[64] in workspace
// ---------------------------------------------------------------------------
__global__ void zero_kernel(float* __restrict__ buf, long n) {
  long i = (long)blockIdx.x * blockDim.x + threadIdx.x;
  long stride = (long)gridDim.x * blockDim.x;
  for (; i < n; i += stride) buf[i] = 0.0f;
}

// ---------------------------------------------------------------------------
// Edge kernel: e = relu([edge|node_r|node_s] @ We1 + be1) @ We2 + be2
// + scatter-add into agg[receivers], + partial sum_e
// 6 waves/WG. LDS (floats): W1p 24576 | W2p 8192 | be1 128 | be2 64 | sum 64
//                           | hbuf 6*1024   -> 156,672 bytes
// ---------------------------------------------------------------------------
#define E_WAVES 6
#define E_THREADS (E_WAVES * 32)
#define E_OFF_W1 0
#define E_OFF_W2 24576
#define E_OFF_B1 32768
#define E_OFF_B2 32896
#define E_OFF_SUM 32960
#define E_OFF_H 33024
#define E_LDS_FLOATS (E_OFF_H + E_WAVES * 1024)
#define E_LDS_BYTES (E_LDS_FLOATS * 4)

__global__ void __launch_bounds__(E_THREADS) edge_kernel(
    const float* __restrict__ nodes, const float* __restrict__ edges,
    const int* __restrict__ senders, const int* __restrict__ receivers,
    const float* __restrict__ We1, const float* __restrict__ be1,
    const float* __restrict__ We2, const float* __restrict__ be2,
    float* __restrict__ e_out, float* __restrict__ agg,
    float* __restrict__ sum_e) {
  extern __shared__ float smem[];
  const int tid = threadIdx.x;
  // Swizzled cooperative weight load (one-time per WG; reads fully coalesced)
  for (int i = tid; i < 192 * HH; i += E_THREADS) {
    const int k = i >> 7, c = i & 127;
    smem[E_OFF_W1 + wswz(k, c, 256)] = We1[i];
  }
  for (int i = tid; i < HH * DD; i += E_THREADS) {
    const int k = i >> 6, c = i & 63;
    smem[E_OFF_W2 + wswz(k, c, 128)] = We2[i];
  }
  if (tid < 128) smem[E_OFF_B1 + tid] = be1[tid];
  if (tid < 64) smem[E_OFF_B2 + tid] = be2[tid];
  if (tid < 64) smem[E_OFF_SUM + tid] = 0.0f;
  __syncthreads();

  const int wave = tid >> 5;
  const int lane = tid & 31;
  const int m = lane & 15;    // tile row (A) / column (B,C)
  const int half = lane >> 4; // K-pair selector
  const int xh = half << 5;   // bank-swizzle XOR for B fragments
  float* hbuf = &smem[E_OFF_H + wave * 1024]; // swizzled [16][64]

  const int nTiles = N_EDGES / 16;
  const int gwave = blockIdx.x * E_WAVES + wave;
  const int nwaves = gridDim.x * E_WAVES;

  float sacc[4] = {0.f, 0.f, 0.f, 0.f};

  for (int t = gwave; t < nTiles; t += nwaves) {
    const int e0 = t * 16;
    const int er = e0 + m;
    const float* pe = edges + (size_t)er * DD;
    const float* pr = nodes + (size_t)receivers[er] * DD;
    const float* ps = nodes + (size_t)senders[er] * DD;

    // ---- layer 1: h[16,128] = x[16,192] @ We1 ----
    v8f hacc[8] = {};
    for (int kk = 0; kk < 48; ++kk) {
      const int k0 = kk * 4 + half * 2;
      const float* p = (k0 < 64) ? (pe + k0)
                       : (k0 < 128) ? (pr + (k0 - 64))
                                    : (ps + (k0 - 128));
      v2f a = *(const v2f*)p; // fused gather of concat(edge,node_r,node_s)
      const int wbase = E_OFF_W1 + (2 * kk + half) * 256;
#pragma unroll
      for (int nt = 0; nt < 8; ++nt) {
        v2f b = *(const v2f*)&smem[wbase + (((nt * 16 + m) << 1) ^ xh)];
        hacc[nt] = wmma_f32(a, b, hacc[nt]);
      }
    }

    // ---- layer 2: e[16,64] = relu(h+b1) @ We2, via swizzled hbuf, 2 halves --
    v8f eacc[4] = {};
#pragma unroll
    for (int hk = 0; hk < 2; ++hk) {
      asm volatile("s_wait_dscnt 0" ::: "memory"); // hbuf reuse fence
#pragma unroll
      for (int nt = 0; nt < 4; ++nt) {
        const int colh = (hk * 4 + nt) * 16 + m;
        const float bias = smem[E_OFF_B1 + colh];
#pragma unroll
        for (int v = 0; v < 8; ++v) {
          float hv = hacc[hk * 4 + nt][v] + bias;
          hv = hv > 0.f ? hv : 0.f;
          hbuf[hswz(half * 8 + v, nt * 16 + m)] = hv;
        }
      }
      asm volatile("s_wait_dscnt 0" ::: "memory");
#pragma unroll
      for (int kk2 = 0; kk2 < 16; ++kk2) {
        const int kl = kk2 * 4 + half * 2;
        v2f a2 = *(const v2f*)&hbuf[m * 64 + (kl ^ (m << 2))];
        const int wbase = E_OFF_W2 + (hk * 32 + 2 * kk2 + half) * 128;
#pragma unroll
        for (int nt2 = 0; nt2 < 4; ++nt2) {
          v2f b2 = *(const v2f*)&smem[wbase + (((nt2 * 16 + m) << 1) ^ xh)];
          eacc[nt2] = wmma_f32(a2, b2, eacc[nt2]);
        }
      }
    }

    // ---- epilogue: bias, store e, scatter-add to agg, sum_e partials ----
#pragma unroll
    for (int v = 0; v < 8; ++v) {
      const int erow = e0 + half * 8 + v;
      const int rcv = receivers[erow];
#pragma unroll
      for (int nt2 = 0; nt2 < 4; ++nt2) {
        const int col = nt2 * 16 + m;
        const float val = eacc[nt2][v] + smem[E_OFF_B2 + col];
        e_out[(size_t)erow * DD + col] = val;
        unsafeAtomicAdd(&agg[(size_t)rcv * DD + col], val);
        sacc[nt2] += val;
      }
    }
  }

#pragma unroll
  for (int nt2 = 0; nt2 < 4; ++nt2)
    atomicAdd(&smem[E_OFF_SUM + nt2 * 16 + m], sacc[nt2]);
  __syncthreads();
  if (tid < 64) unsafeAtomicAdd(&sum_e[tid], smem[E_OFF_SUM + tid]);
}

// ---------------------------------------------------------------------------
// Node kernel: n = relu([agg|node] @ Wn1 + bn1) @ Wn2 + bn2, + partial sum_n
// 8 waves/WG. LDS (floats): W1p 16384 | W2p 8192 | bn1 128 | bn2 64 | sum 64
//                           | hbuf 8*1024   -> 132,096 bytes
// ---------------------------------------------------------------------------
#define V_WAVES 8
#define V_THREADS (V_WAVES * 32)
#define V_OFF_W1 0
#define V_OFF_W2 16384
#define V_OFF_B1 24576
#define V_OFF_B2 24704
#define V_OFF_SUM 24768
#define V_OFF_H 24832
#define V_LDS_FLOATS (V_OFF_H + V_WAVES * 1024)
#define V_LDS_BYTES (V_LDS_FLOATS * 4)

__global__ void __launch_bounds__(V_THREADS) node_kernel(
    const float* __restrict__ nodes, const float* __restrict__ agg,
    const float* __restrict__ Wn1, const float* __restrict__ bn1,
    const float* __restrict__ Wn2, const float* __restrict__ bn2,
    float* __restrict__ n_out, float* __restrict__ sum_n) {
  extern __shared__ float smem[];
  const int tid = threadIdx.x;
  for (int i = tid; i < HH * HH; i += V_THREADS) {
    const int k = i >> 7, c = i & 127;
    smem[V_OFF_W1 + wswz(k, c, 256)] = Wn1[i];
  }
  for (int i = tid; i < HH * DD; i += V_THREADS) {
    const int k = i >> 6, c = i & 63;
    smem[V_OFF_W2 + wswz(k, c, 128)] = Wn2[i];
  }
  if (tid < 128) smem[V_OFF_B1 + tid] = bn1[tid];
  if (tid < 64) smem[V_OFF_B2 + tid] = bn2[tid];
  if (tid < 64) smem[V_OFF_SUM + tid] = 0.0f;
  __syncthreads();

  const int wave = tid >> 5;
  const int lane = tid & 31;
  const int m = lane & 15;
  const int half = lane >> 4;
  const int xh = half << 5;
  float* hbuf = &smem[V_OFF_H + wave * 1024];

  const int nTiles = N_NODES / 16;
  const int gwave = blockIdx.x * V_WAVES + wave;
  const int nwaves = gridDim.x * V_WAVES;

  float sacc[4] = {0.f, 0.f, 0.f, 0.f};

  for (int t = gwave; t < nTiles; t += nwaves) {
    const int n0 = t * 16;
    const int nr = n0 + m;
    const float* pa = agg + (size_t)nr * DD;
    const float* pn = nodes + (size_t)nr * DD;

    v8f hacc[8] = {};
    for (int kk = 0; kk < 32; ++kk) {
      const int k0 = kk * 4 + half * 2;
      const float* p = (k0 < 64) ? (pa + k0) : (pn + (k0 - 64));
      v2f a = *(const v2f*)p;
      const int wbase = V_OFF_W1 + (2 * kk + half) * 256;
#pragma unroll
      for (int nt = 0; nt < 8; ++nt) {
        v2f b = *(const v2f*)&smem[wbase + (((nt * 16 + m) << 1) ^ xh)];
        hacc[nt] = wmma_f32(a, b, hacc[nt]);
      }
    }

    v8f nacc[4] = {};
#pragma unroll
    for (int hk = 0; hk < 2; ++hk) {
      asm volatile("s_wait_dscnt 0" ::: "memory");
#pragma unroll
      for (int nt = 0; nt < 4; ++nt) {
        const int colh = (hk * 4 + nt) * 16 + m;
        const float bias = smem[V_OFF_B1 + colh];
#pragma unroll
        for (int v = 0; v < 8; ++v) {
          float hv = hacc[hk * 4 + nt][v] + bias;
          hv = hv > 0.f ? hv : 0.f;
          hbuf[hswz(half * 8 + v, nt * 16 + m)] = hv;
        }
      }
      asm volatile("s_wait_dscnt 0" ::: "memory");
#pragma unroll
      for (int kk2 = 0; kk2 < 16; ++kk2) {
        const int kl = kk2 * 4 + half * 2;
        v2f a2 = *(const v2f*)&hbuf[m * 64 + (kl ^ (m << 2))];
        const int wbase = V_OFF_W2 + (hk * 32 + 2 * kk2 + half) * 128;
#pragma unroll
        for (int nt2 = 0; nt2 < 4; ++nt2) {
          v2f b2 = *(const v2f*)&smem[wbase + (((nt2 * 16 + m) << 1) ^ xh)];
          nacc[nt2] = wmma_f32(a2, b2, nacc[nt2]);
        }
      }
    }

#pragma unroll
    for (int v = 0; v < 8; ++v) {
      const int nrow = n0 + half * 8 + v;
#pragma unroll
      for (int nt2 = 0; nt2 < 4; ++nt2) {
        const int col = nt2 * 16 + m;
        const float val = nacc[nt2][v] + smem[V_OFF_B2 + col];
        n_out[(size_t)nrow * DD + col] = val;
        sacc[nt2] += val;
      }
    }
  }

#pragma unroll
  for (int nt2 = 0; nt2 < 4; ++nt2)
    atomicAdd(&smem[V_OFF_SUM + nt2 * 16 + m], sacc[nt2]);
  __syncthreads();
  if (tid < 64) unsafeAtomicAdd(&sum_n[tid], smem[V_OFF_SUM + tid]);
}

// ---------------------------------------------------------------------------
// Global kernel: g = relu([sum_n|sum_e] @ Wg1 + bg1) @ Wg2 + bg2  (tiny)
// ---------------------------------------------------------------------------
__global__ void __launch_bounds__(128) global_kernel(
    const float* __restrict__ sum_n, const float* __restrict__ sum_e,
    const float* __restrict__ Wg1, const float* __restrict__ bg1,
    const float* __restrict__ Wg2, const float* __restrict__ bg2,
    float* __restrict__ g_out) {
  __shared__ float gin[128];
  __shared__ float h[128];
  const int tid = threadIdx.x;
  gin[tid] = (tid < 64) ? sum_n[tid] : sum_e[tid - 64];
  __syncthreads();
  float acc = bg1[tid];
  for (int k = 0; k < 128; ++k) acc += gin[k] * Wg1[k * HH + tid];
  h[tid] = acc > 0.f ? acc : 0.f;
  __syncthreads();
  if (tid < 64) {
    float a2 = bg2[tid];
    for (int k = 0; k < 128; ++k) a2 += h[k] * Wg2[k * DD + tid];
    g_out[tid] = a2;
  }
}

// ---------------------------------------------------------------------------
extern "C" void kernel_launch(void* const* d_in, const int* in_sizes, int n_in,
                              void* d_out, int out_size, void* d_ws,
                              size_t ws_size, hipStream_t stream) {
  const float* nodes = (const float*)d_in[0];
  const float* edges = (const float*)d_in[1];
  const int* senders = (const int*)d_in[2];
  const int* receivers = (const int*)d_in[3];
  const float* We1 = (const float*)d_in[4];
  const float* be1 = (const float*)d_in[5];
  const float* We2 = (const float*)d_in[6];
  const float* be2 = (const float*)d_in[7];
  const float* Wn1 = (const float*)d_in[8];
  const float* bn1 = (const float*)d_in[9];
  const float* Wn2 = (const float*)d_in[10];
  const float* bn2 = (const float*)d_in[11];
  const float* Wg1 = (const float*)d_in[12];
  const float* bg1 = (const float*)d_in[13];
  const float* Wg2 = (const float*)d_in[14];
  const float* bg2 = (const float*)d_in[15];

  float* out = (float*)d_out;
  float* n_out = out;                                               // [N,64]
  float* e_out = out + (size_t)N_NODES * DD;                        // [E,64]
  float* g_out = out + (size_t)N_NODES * DD + (size_t)N_EDGES * DD; // [64]

  float* agg = (float*)d_ws;                 // [N,64]
  float* sum_n = agg + (size_t)N_NODES * DD; // [64]
  float* sum_e = sum_n + 64;                 // [64]

  hipFuncSetAttribute((const void*)edge_kernel,
                      hipFuncAttributeMaxDynamicSharedMemorySize, E_LDS_BYTES);
  hipFuncSetAttribute((const void*)node_kernel,
                      hipFuncAttributeMaxDynamicSharedMemorySize, V_LDS_BYTES);

  const long zero_n = (long)N_NODES * DD + 128;
  zero_kernel<<<2048, 256, 0, stream>>>(agg, zero_n);

  edge_kernel<<<800, E_THREADS, E_LDS_BYTES, stream>>>(
      nodes, edges, senders, receivers, We1, be1, We2, be2, e_out, agg, sum_e);

  node_kernel<<<256, V_THREADS, V_LDS_BYTES, stream>>>(
      nodes, agg, Wn1, bn1, Wn2, bn2, n_out, sum_n);

  global_kernel<<<1, 128, 0, stream>>>(sum_n, sum_e, Wg1, bg1, Wg2, bg2, g_out);
}